// DynamicConv1dTBC_60138132078904
// MI455X (gfx1250) — compile-verified
//
#include <hip/hip_runtime.h>
#include <hip/hip_bf16.h>
#include <math.h>
#include <stdint.h>

typedef __attribute__((ext_vector_type(2))) float v2f;
typedef __attribute__((ext_vector_type(4))) float v4f;
typedef __attribute__((ext_vector_type(8))) float v8f;

#define T_DIM 2048
#define B_DIM 8
#define C_DIM 1024
#define H_DIM 16
#define K_TAPS 15
#define M_DIM (T_DIM * B_DIM)   // 16384

// --- CDNA5 async global->LDS copy (tracked by ASYNCcnt), GVS addressing:
//     global address = SGPR base + VGPR byte offset; 16 bytes land in LDS at lds_off
__device__ __forceinline__ void async_ld_b128(uint32_t lds_off, uint32_t gbyte_off,
                                              uint64_t base) {
    asm volatile("global_load_async_to_lds_b128 %0, %1, %2 offset:0"
                 :: "v"(lds_off), "v"(gbyte_off), "s"(base)
                 : "memory");
}
#define WAIT_ASYNCCNT(n) asm volatile("s_wait_asynccnt %0" :: "n"(n) : "memory")

// LDS byte offset of a __shared__ object (generic shared VA: low 32 bits = LDS offset)
__device__ __forceinline__ uint32_t lds_off_of(const void* p) {
    return (uint32_t)(uintptr_t)p;
}

// ---------------------------------------------------------------------------
// fp32 WMMA GEMM:  out[M,N] = A[M,K] @ W[N,K]^T (+bias)
// 256 threads (8 waves), tile 64(M) x 128(N), K-chunks of 32,
// async double-buffered LDS staging (ASYNCcnt), fp32 WMMA 16x16x4.
// mode 0: plain store (stride N, optional bias, N edge-guarded)
// mode 1: GLU epilogue — block tile == one head's 128 cols of gate_pre;
//         writes gate = a * sigmoid(b) into out[m, h*64 + j], stride 1024.
// ---------------------------------------------------------------------------
__global__ __launch_bounds__(256) void gemm_f32_wmma(
    const float* __restrict__ A, const float* __restrict__ W,
    const float* __restrict__ bias, float* __restrict__ out,
    int M, int N, int K, int mode)
{
    __shared__ union {
        struct {
            float As[2][64 * 34];    // 2 x  8704 B  (padded stride 34)
            float Bs[2][128 * 34];   // 2 x 17408 B
        } stage;                     // 52 KB live during K-loop
        float Cs[64 * 132];          // 33 KB live after K-loop (union overlap)
    } sm;

    const int tid   = threadIdx.x;
    const int lane  = tid & 31;          // wave32
    const int wid   = tid >> 5;          // 0..7
    const int mi    = wid >> 1;          // 0..3
    const int ni    = wid & 1;           // 0..1
    const int lrow  = lane & 15;
    const int khalf = (lane >> 4) << 1;  // lanes 0-15: K {k,k+1}; 16-31: {k+2,k+3}
    const int rbase = (lane >> 4) << 3;  // C rows: lanes 0-15 -> M 0..7; 16-31 -> 8..15

    const int n0 = blockIdx.x * 128;
    const int m0 = blockIdx.y * 64;

    const uint64_t Abase = (uint64_t)(uintptr_t)A;
    const uint64_t Wbase = (uint64_t)(uintptr_t)W;

    // per-thread staging coordinates (uniform 6 async b128 per thread per chunk)
    const int sr = tid >> 3;             // 0..31
    const int sc = (tid & 7) << 2;       // 0,4,..,28

    // issue one K-chunk (32 wide) into stage buffer `buf`
    auto issue_chunk = [&](int kc, int buf) {
#pragma unroll
        for (int i = 0; i < 2; ++i) {     // A tile: 64 x 32
            int r = sr + i * 32;
            async_ld_b128(lds_off_of(&sm.stage.As[buf][r * 34 + sc]),
                          (uint32_t)((((size_t)(m0 + r)) * K + kc + sc) * 4), Abase);
        }
#pragma unroll
        for (int i = 0; i < 4; ++i) {     // B tile: 128 x 32 (clamp OOB rows: the
            int r  = sr + i * 32;         // garbage only feeds never-stored columns)
            int ng = n0 + r; if (ng >= N) ng = N - 1;
            async_ld_b128(lds_off_of(&sm.stage.Bs[buf][r * 34 + sc]),
                          (uint32_t)((((size_t)ng) * K + kc + sc) * 4), Wbase);
        }
    };

    v8f acc[4];
#pragma unroll
    for (int s = 0; s < 4; ++s) acc[s] = (v8f){0.f,0.f,0.f,0.f,0.f,0.f,0.f,0.f};

    const int nch = K >> 5;              // K/32 chunks
    issue_chunk(0, 0);

    for (int ic = 0; ic < nch; ++ic) {
        const int buf = ic & 1;
        if (ic + 1 < nch) {
            issue_chunk((ic + 1) << 5, buf ^ 1);  // prefetch next chunk
            WAIT_ASYNCCNT(6);                     // chunk ic landed; next in flight
        } else {
            WAIT_ASYNCCNT(0);
        }
        __syncthreads();                          // all waves' chunk-ic data visible

        const float* Asb = sm.stage.As[buf];
        const float* Bsb = sm.stage.Bs[buf];
#pragma unroll
        for (int kk = 0; kk < 32; kk += 4) {
            v2f a = *(const v2f*)&Asb[(mi * 16 + lrow) * 34 + kk + khalf];
#pragma unroll
            for (int s = 0; s < 4; ++s) {
                v2f b = *(const v2f*)&Bsb[(ni * 64 + s * 16 + lrow) * 34 + kk + khalf];
                acc[s] = __builtin_amdgcn_wmma_f32_16x16x4_f32(
                    false, a, false, b, (short)0, acc[s], false, false);
            }
        }
        __syncthreads();                          // done reading buf -> reusable
    }

    // --- spill accumulators to LDS (union: stage buffers dead now) ---
#pragma unroll
    for (int s = 0; s < 4; ++s)
#pragma unroll
        for (int r = 0; r < 8; ++r)
            sm.Cs[(mi * 16 + rbase + r) * 132 + ni * 64 + s * 16 + lrow] = acc[s][r];
    __syncthreads();

    if (mode == 0) {
        // plain: 64 rows x 128 cols, float4 coalesced stores, N guard
#pragma unroll
        for (int rr = 0; rr < 8; ++rr) {
            int row  = wid + rr * 8;
            int c4   = lane << 2;
            int gcol = n0 + c4;
            if (gcol >= N) continue;
            v4f v = *(const v4f*)&sm.Cs[row * 132 + c4];
            float* op = &out[(size_t)(m0 + row) * N + gcol];
            if (gcol + 3 < N) {
                if (bias) {
                    v.x += bias[gcol]; v.y += bias[gcol + 1];
                    v.z += bias[gcol + 2]; v.w += bias[gcol + 3];
                }
                *(v4f*)op = v;
            } else {
#pragma unroll
                for (int i = 0; i < 4; ++i)
                    if (gcol + i < N) op[i] = v[i] + (bias ? bias[gcol + i] : 0.f);
            }
        }
    } else {
        // GLU: block == head h; cols [0,64) = a, [64,128) = b; gate = a*sigmoid(b)
        int h = n0 >> 7;
#pragma unroll
        for (int rr = 0; rr < 4; ++rr) {
            int row = (tid >> 4) + rr * 16;
            int j   = (tid & 15) << 2;
            v4f va  = *(const v4f*)&sm.Cs[row * 132 + j];
            v4f vb  = *(const v4f*)&sm.Cs[row * 132 + 64 + j];
            v4f g;
#pragma unroll
            for (int i = 0; i < 4; ++i) {
                float a  = va[i] + bias[n0 + j + i];
                float bg = vb[i] + bias[n0 + 64 + j + i];
                g[i] = a * (1.f / (1.f + __expf(-bg)));
            }
            *(v4f*)&out[(size_t)(m0 + row) * C_DIM + h * 64 + j] = g;
        }
    }
}

// ---------------------------------------------------------------------------
// Softmax over K=15 taps, in place. One thread per (m, h); rows = M*H.
// ---------------------------------------------------------------------------
__global__ __launch_bounds__(256) void softmax15_kernel(float* __restrict__ w, int rows)
{
    int idx = blockIdx.x * blockDim.x + threadIdx.x;
    if (idx >= rows) return;
    int m = idx >> 4, h = idx & 15;
    float* p = w + (size_t)m * (H_DIM * K_TAPS) + h * K_TAPS;
    float v[K_TAPS], mx = -INFINITY;
#pragma unroll
    for (int k = 0; k < K_TAPS; ++k) { v[k] = p[k]; mx = fmaxf(mx, v[k]); }
    float sum = 0.f;
#pragma unroll
    for (int k = 0; k < K_TAPS; ++k) { v[k] = __expf(v[k] - mx); sum += v[k]; }
    float inv = 1.f / sum;
#pragma unroll
    for (int k = 0; k < K_TAPS; ++k) p[k] = v[k] * inv;
}

// ---------------------------------------------------------------------------
// Causal dynamic depthwise conv + gate, in place on the gate buffer.
// One thread per (t,b, 4 channels); float4 coalesced; x served from L2.
//   gio[m, c4..c4+3] = gate * sum_k w[m, h*15+k] * x[t-14+k, b, c4..c4+3]
// ---------------------------------------------------------------------------
__global__ __launch_bounds__(256) void dynconv_kernel(
    const float* __restrict__ x, const float* __restrict__ wsm,
    float* __restrict__ gio)
{
    int idx = blockIdx.x * blockDim.x + threadIdx.x;   // over M * 256
    int c4 = (idx & 255) << 2;
    int m  = idx >> 8;
    int t  = m >> 3;                                    // B_DIM = 8
    int b  = m & (B_DIM - 1);
    int h  = c4 >> 6;

    const float* wr = wsm + (size_t)m * (H_DIM * K_TAPS) + h * K_TAPS;
    float wk[K_TAPS];
#pragma unroll
    for (int k = 0; k < K_TAPS; ++k) wk[k] = wr[k];

    v4f acc = (v4f){0.f, 0.f, 0.f, 0.f};
#pragma unroll
    for (int k = 0; k < K_TAPS; ++k) {
        int tt = t + k - (K_TAPS - 1);
        if (tt >= 0) {
            v4f xv = *(const v4f*)&x[((size_t)tt * B_DIM + b) * C_DIM + c4];
            acc += xv * wk[k];
        }
    }
    v4f* gp = (v4f*)&gio[(size_t)m * C_DIM + c4];
    *gp = acc * (*gp);
}

// ---------------------------------------------------------------------------
extern "C" void kernel_launch(void* const* d_in, const int* in_sizes, int n_in,
                              void* d_out, int out_size, void* d_ws, size_t ws_size,
                              hipStream_t stream)
{
    const float* x        = (const float*)d_in[0];   // [T,B,C]
    const float* w_weight = (const float*)d_in[1];   // [240, C]
    const float* glu_w    = (const float*)d_in[2];   // [2C, C]
    const float* glu_b    = (const float*)d_in[3];   // [2C]
    const float* out_w    = (const float*)d_in[4];   // [C, C]
    const float* out_b    = (const float*)d_in[5];   // [C]
    float* outp = (float*)d_out;

    // workspace: gate/y buffer [M,C] (64 MB) then conv-weight buffer [M,240]
    float* gbuf = (float*)d_ws;
    float* wbuf = gbuf + (size_t)M_DIM * C_DIM;

    dim3 blk(256);

    // 1) GLU GEMM + fused gate epilogue  (N = 2C = 2048, 16 head-blocks)
    gemm_f32_wmma<<<dim3(2 * C_DIM / 128, M_DIM / 64), blk, 0, stream>>>(
        x, glu_w, glu_b, gbuf, M_DIM, 2 * C_DIM, C_DIM, 1);

    // 2) conv-weight logits GEMM (N = 240, edge-guarded)
    gemm_f32_wmma<<<dim3((H_DIM * K_TAPS + 127) / 128, M_DIM / 64), blk, 0, stream>>>(
        x, w_weight, nullptr, wbuf, M_DIM, H_DIM * K_TAPS, C_DIM, 0);

    // 3) softmax over 15 taps, in place
    softmax15_kernel<<<(M_DIM * H_DIM + 255) / 256, blk, 0, stream>>>(
        wbuf, M_DIM * H_DIM);

    // 4) dynamic causal conv * gate, in place (gbuf becomes y)
    dynconv_kernel<<<(M_DIM * (C_DIM / 4)) / 256, blk, 0, stream>>>(x, wbuf, gbuf);

    // 5) output projection GEMM + bias -> d_out
    gemm_f32_wmma<<<dim3(C_DIM / 128, M_DIM / 64), blk, 0, stream>>>(
        gbuf, out_w, out_b, outp, M_DIM, C_DIM, C_DIM, 0);
}